// SpatialAbstraction_4398046511722
// MI455X (gfx1250) — compile-verified
//
#include <hip/hip_runtime.h>

#define B_ 8
#define N_ 8192
#define D_ 64
#define S_ 2048
#define K_ 32
#define R_ (B_*S_*K_)          // 524288 rows
#define RADIUS_ 0.5f
#define EPS_ 1e-5f

typedef __attribute__((ext_vector_type(2))) float v2f;
typedef __attribute__((ext_vector_type(8))) float v8f;

// ---------------------------------------------------------------------------
// 1) Furthest point sampling: one block per batch; xyz + dist live in LDS
//    (96KB + 32KB of the 320KB WGP LDS). 2048 sequential argmax rounds.
// ---------------------------------------------------------------------------
__global__ __launch_bounds__(1024) void fps_kernel(
    const float* __restrict__ xyz,      // [B,3,N]
    const float* __restrict__ points,   // [B,D,N]
    float* __restrict__ new_xyz_t,      // ws [B,S,3]
    float* __restrict__ new_feat,       // ws [B,S,D]
    float* __restrict__ out,            // d_out base (new_xyz written twice)
    int* __restrict__ fps_idx_out)      // d_out fps region [B,S]
{
    extern __shared__ float sm[];
    float* sx    = sm;
    float* sy    = sm + N_;
    float* sz    = sm + 2 * N_;
    float* sdist = sm + 3 * N_;
    float* rval  = sm + 4 * N_;               // [1024]
    int*   ridx  = (int*)(rval + 1024);       // [1024]
    __shared__ int sfar;

    const int b = blockIdx.x;
    const int t = threadIdx.x;
    const float* xb = xyz + (size_t)b * 3 * N_;

    for (int i = t; i < N_; i += 1024) {
        sx[i] = xb[i];
        sy[i] = xb[N_ + i];
        sz[i] = xb[2 * N_ + i];
        sdist[i] = 1e10f;
    }
    __syncthreads();

    int far = 0;
    for (int it = 0; it < S_; ++it) {
        const float px = sx[far], py = sy[far], pz = sz[far];
        if (t == 0) {
            fps_idx_out[b * S_ + it] = far;
            float* nxt = new_xyz_t + ((size_t)b * S_ + it) * 3;
            nxt[0] = px; nxt[1] = py; nxt[2] = pz;
            const size_t o0 = (size_t)b * 3 * S_;
            out[o0 + 0 * S_ + it] = px;
            out[o0 + 1 * S_ + it] = py;
            out[o0 + 2 * S_ + it] = pz;
            float* out1 = out + (size_t)B_ * 3 * S_;   // duplicate new_xyz output
            out1[o0 + 0 * S_ + it] = px;
            out1[o0 + 1 * S_ + it] = py;
            out1[o0 + 2 * S_ + it] = pz;
        }
        if (t < D_) {   // gather center features (points_t at 'far')
            new_feat[((size_t)b * S_ + it) * D_ + t] =
                points[((size_t)b * D_ + t) * N_ + far];
        }
        // update min-dist and per-thread argmax (first-index tie-break)
        float bv = -1.0f; int bi = 0;
        for (int i = t; i < N_; i += 1024) {
            const float dx = sx[i] - px, dy = sy[i] - py, dz = sz[i] - pz;
            const float d = dx * dx + dy * dy + dz * dz;
            const float dd = fminf(sdist[i], d);
            sdist[i] = dd;
            if (dd > bv) { bv = dd; bi = i; }
        }
        rval[t] = bv; ridx[t] = bi;
        __syncthreads();
        for (int s2 = 512; s2 > 0; s2 >>= 1) {
            if (t < s2) {
                const float ov = rval[t + s2]; const int oi = ridx[t + s2];
                if (ov > rval[t] || (ov == rval[t] && oi < ridx[t])) {
                    rval[t] = ov; ridx[t] = oi;
                }
            }
            __syncthreads();
        }
        if (t == 0) sfar = ridx[0];
        __syncthreads();
        far = sfar;
    }
}

// ---------------------------------------------------------------------------
// 2) Ball query: one wave32 per center; ballot+popcount keeps the
//    "first nsample in ascending index order, pad with first" semantics.
// ---------------------------------------------------------------------------
__global__ __launch_bounds__(256) void ballquery_kernel(
    const float* __restrict__ xyz,        // [B,3,N]
    const float* __restrict__ new_xyz_t,  // [B,S,3]
    int* __restrict__ idx)                // [B,S,K]
{
    const int gw   = (blockIdx.x * 256 + threadIdx.x) >> 5;  // center id b*S+s
    const int lane = threadIdx.x & 31;
    const int b = gw >> 11;                                  // S_ = 2048
    const float* xb = xyz + (size_t)b * 3 * N_;
    const float cx = new_xyz_t[(size_t)gw * 3 + 0];
    const float cy = new_xyz_t[(size_t)gw * 3 + 1];
    const float cz = new_xyz_t[(size_t)gw * 3 + 2];
    int* ob = idx + (size_t)gw * K_;
    const float r2 = RADIUS_ * RADIUS_;

    int cnt = 0;
    int firstIdx = 0;
    for (int base = 0; base < N_ && cnt < K_; base += 32) {
        const int j = base + lane;
        const float dx = xb[j] - cx, dy = xb[N_ + j] - cy, dz = xb[2 * N_ + j] - cz;
        const bool in = (dx * dx + dy * dy + dz * dz) <= r2;
        const unsigned mask = __builtin_amdgcn_ballot_w32(in);
        if (mask) {
            if (cnt == 0) firstIdx = base + __builtin_ctz(mask);
            const int rank = cnt + __builtin_popcount(mask & ((1u << lane) - 1u));
            if (in && rank < K_) ob[rank] = j;
            cnt += __builtin_popcount(mask);
        }
    }
    if (cnt > K_) cnt = K_;
    if (lane >= cnt) ob[lane] = firstIdx;    // first-found fill
}

// ---------------------------------------------------------------------------
// WMMA fp32 layout helpers (V_WMMA_F32_16X16X4_F32):
//   A[M=lane%16][K = vgpr + 2*(lane/16)], B[K = vgpr + 2*(lane/16)][N=lane%16]
//   D[M = vgpr + 8*(lane/16)][N = lane%16]
// ---------------------------------------------------------------------------

// 3) Layer-0 GEMM: fused gather/concat of [rel_xyz | grouped_pts | center_rep]
//    (131 -> padded 132) x W0^T -> feat1 [R,64], plus per-channel sum/sumsq.
__global__ __launch_bounds__(256) void gemm0_kernel(
    const float* __restrict__ xyz,       // [B,3,N]
    const float* __restrict__ points,    // [B,D,N]
    const float* __restrict__ new_xyz_t, // [B,S,3]
    const float* __restrict__ new_feat,  // [B,S,D]
    const int* __restrict__ idx,         // [R] (== [B,S,K] flat)
    const float* __restrict__ W0,        // [64,131]
    float* __restrict__ Y,               // feat1 [R,64]
    float* __restrict__ stat)            // [sum 0..63, sumsq 128..191]
{
    extern __shared__ float sm[];
    float* A    = sm;                       // [128][133] (pad for bank spread)
    float* Bm   = sm + 128 * 133;           // [132][64]  B[c][o] = W0[o][c]
    float* ssum = Bm + 132 * 64;            // [64]
    float* ssq  = ssum + 64;                // [64]
    int*   jrow = (int*)(ssq + 64);         // [128]

    const int t = threadIdx.x;
    const int rowBase = blockIdx.x * 128;

    if (t < 64) { ssum[t] = 0.f; ssq[t] = 0.f; }
    if (t < 128) jrow[t] = idx[(size_t)rowBase + t];
    for (int e = t; e < 132 * 64; e += 256) {
        const int c = e >> 6, o = e & 63;
        Bm[c * 64 + o] = (c < 131) ? W0[o * 131 + c] : 0.f;
    }
    __syncthreads();

    for (int e = t; e < 128 * 132; e += 256) {
        const int r = e / 132, c = e % 132;
        const int gr = rowBase + r;
        const int bs = gr >> 5;             // b*S+s
        const int b  = bs >> 11;
        float v = 0.f;
        if (c < 131) {
            const int j = jrow[r];
            if (c < 3)
                v = xyz[((size_t)b * 3 + c) * N_ + j] - new_xyz_t[(size_t)bs * 3 + c];
            else if (c < 3 + D_)
                v = points[((size_t)b * D_ + (c - 3)) * N_ + j];
            else
                v = new_feat[(size_t)bs * D_ + (c - 3 - D_)];
        }
        A[r * 133 + c] = v;
    }
    __syncthreads();

    const int lane = t & 31, wave = t >> 5;
    const int lm = lane & 15, lh = lane >> 4;
    v8f acc[4];
    #pragma unroll
    for (int nt = 0; nt < 4; ++nt)
        #pragma unroll
        for (int v = 0; v < 8; ++v) acc[nt][v] = 0.f;

    const float* Arow = A + (wave * 16 + lm) * 133;
    for (int k0 = 0; k0 < 132; k0 += 4) {
        v2f a;
        a.x = Arow[k0 + 2 * lh];
        a.y = Arow[k0 + 2 * lh + 1];
        #pragma unroll
        for (int nt = 0; nt < 4; ++nt) {
            v2f bb;
            bb.x = Bm[(k0 + 2 * lh) * 64 + nt * 16 + lm];
            bb.y = Bm[(k0 + 2 * lh + 1) * 64 + nt * 16 + lm];
            acc[nt] = __builtin_amdgcn_wmma_f32_16x16x4_f32(
                false, a, false, bb, (short)0, acc[nt], false, false);
        }
    }

    #pragma unroll
    for (int nt = 0; nt < 4; ++nt) {
        const int col = nt * 16 + lm;
        float s_ = 0.f, q_ = 0.f;
        #pragma unroll
        for (int v = 0; v < 8; ++v) {
            const float y = acc[nt][v];
            const int row = rowBase + wave * 16 + v + 8 * lh;
            Y[(size_t)row * 64 + col] = y;
            s_ += y; q_ += y * y;
        }
        atomicAdd(&ssum[col], s_);
        atomicAdd(&ssq[col], q_);
    }
    __syncthreads();
    if (t < 64) {
        atomicAdd(&stat[t], ssum[t]);
        atomicAdd(&stat[128 + t], ssq[t]);
    }
}

// 4) Mid GEMM: BN(prev stats)+ReLU applied on load, Y = act(X) @ W^T, + stats.
template <int CIN, int COUT>
__global__ __launch_bounds__(256) void gemm_mid_kernel(
    const float* __restrict__ X,     // [R,CIN] pre-activation
    const float* __restrict__ W,     // [COUT,CIN]
    const float* __restrict__ gma, const float* __restrict__ bta,
    const float* __restrict__ mu,  const float* __restrict__ rstd,
    float* __restrict__ Y,           // [R,COUT]
    float* __restrict__ stat)        // [sum 0..COUT-1, sumsq 128..]
{
    extern __shared__ float sm[];
    float* A    = sm;                         // [128][CIN+1]
    float* Bm   = sm + 128 * (CIN + 1);       // [CIN][COUT]
    float* ssum = Bm + CIN * COUT;            // [COUT]
    float* ssq  = ssum + COUT;                // [COUT]

    const int t = threadIdx.x;
    const int rowBase = blockIdx.x * 128;
    const int NT = COUT / 16;

    if (t < COUT) { ssum[t] = 0.f; ssq[t] = 0.f; }
    for (int e = t; e < CIN * COUT; e += 256) {
        const int c = e / COUT, o = e % COUT;
        Bm[c * COUT + o] = W[o * CIN + c];
    }
    for (int e = t; e < 128 * CIN; e += 256) {
        const int r = e / CIN, c = e % CIN;
        const float x = X[(size_t)(rowBase + r) * CIN + c];
        const float y = gma[c] * (x - mu[c]) * rstd[c] + bta[c];
        A[r * (CIN + 1) + c] = fmaxf(y, 0.f);
    }
    __syncthreads();

    const int lane = t & 31, wave = t >> 5;
    const int lm = lane & 15, lh = lane >> 4;
    v8f acc[COUT / 16];
    #pragma unroll
    for (int nt = 0; nt < COUT / 16; ++nt)
        #pragma unroll
        for (int v = 0; v < 8; ++v) acc[nt][v] = 0.f;

    const float* Arow = A + (wave * 16 + lm) * (CIN + 1);
    for (int k0 = 0; k0 < CIN; k0 += 4) {
        v2f a;
        a.x = Arow[k0 + 2 * lh];
        a.y = Arow[k0 + 2 * lh + 1];
        #pragma unroll
        for (int nt = 0; nt < COUT / 16; ++nt) {
            v2f bb;
            bb.x = Bm[(k0 + 2 * lh) * COUT + nt * 16 + lm];
            bb.y = Bm[(k0 + 2 * lh + 1) * COUT + nt * 16 + lm];
            acc[nt] = __builtin_amdgcn_wmma_f32_16x16x4_f32(
                false, a, false, bb, (short)0, acc[nt], false, false);
        }
    }

    #pragma unroll
    for (int nt = 0; nt < COUT / 16; ++nt) {
        const int col = nt * 16 + lm;
        float s_ = 0.f, q_ = 0.f;
        #pragma unroll
        for (int v = 0; v < 8; ++v) {
            const float y = acc[nt][v];
            const int row = rowBase + wave * 16 + v + 8 * lh;
            Y[(size_t)row * COUT + col] = y;
            s_ += y; q_ += y * y;
        }
        atomicAdd(&ssum[col], s_);
        atomicAdd(&ssq[col], q_);
    }
    __syncthreads();
    if (t < COUT) {
        atomicAdd(&stat[t], ssum[t]);
        atomicAdd(&stat[128 + t], ssq[t]);
    }
    (void)NT;
}

// 5) stats -> (mu, rstd). Layout per layer: sum[128], sumsq[128], mu[128], rstd[128]
__global__ void stats_finalize_kernel(float* stat, int C, float invR) {
    const int t = threadIdx.x;
    if (t < C) {
        const float mu  = stat[t] * invR;
        const float var = stat[128 + t] * invR - mu * mu;
        stat[256 + t] = mu;
        stat[384 + t] = rsqrtf(var + EPS_);
    }
}

// 6) BN2 + ReLU + max over K, stored transposed [B,128,S]
__global__ __launch_bounds__(128) void maxpool_kernel(
    const float* __restrict__ feat3,   // [R,128]
    const float* __restrict__ mu, const float* __restrict__ rstd,
    const float* __restrict__ gma, const float* __restrict__ bta,
    float* __restrict__ out_np)        // [B,128,S]
{
    const int o  = threadIdx.x;        // channel 0..127
    const int bs = blockIdx.x;         // 0..B*S-1
    const float m_ = mu[o], r_ = rstd[o], g_ = gma[o], b_ = bta[o];
    float m = 0.f;                     // relu() >= 0
    for (int k = 0; k < K_; ++k) {
        const float x = feat3[((size_t)bs * K_ + k) * 128 + o];
        const float y = fmaxf(g_ * (x - m_) * r_ + b_, 0.f);
        m = fmaxf(m, y);
    }
    const int b = bs >> 11, s = bs & (S_ - 1);
    out_np[((size_t)b * 128 + o) * S_ + s] = m;
}

// ---------------------------------------------------------------------------
extern "C" void kernel_launch(void* const* d_in, const int* in_sizes, int n_in,
                              void* d_out, int out_size, void* d_ws, size_t ws_size,
                              hipStream_t stream) {
    const float* xyz    = (const float*)d_in[0];
    const float* points = (const float*)d_in[1];
    const float* W0 = (const float*)d_in[2];
    const float* g0 = (const float*)d_in[3];
    const float* b0 = (const float*)d_in[4];
    const float* W1 = (const float*)d_in[5];
    const float* g1 = (const float*)d_in[6];
    const float* b1 = (const float*)d_in[7];
    const float* W2 = (const float*)d_in[8];
    const float* g2 = (const float*)d_in[9];
    const float* b2 = (const float*)d_in[10];

    float* out = (float*)d_out;
    const size_t nxyz = (size_t)B_ * 3 * S_;
    float* out_np  = out + 2 * nxyz;
    int*   out_fps = (int*)(out + 2 * nxyz + (size_t)B_ * 128 * S_);

    float* w = (float*)d_ws;
    float* ws_newxyz = w;           w += (size_t)B_ * S_ * 3;
    float* ws_newfeat = w;          w += (size_t)B_ * S_ * D_;
    int*   ws_idx = (int*)w;        w += (size_t)B_ * S_ * K_;
    float* ws_f1 = w;               w += (size_t)R_ * 64;
    float* ws_f2 = w;               w += (size_t)R_ * 64;
    float* ws_f3 = w;               w += (size_t)R_ * 128;
    float* ws_st0 = w;              w += 512;
    float* ws_st1 = w;              w += 512;
    float* ws_st2 = w;              w += 512;

    hipMemsetAsync(ws_st0, 0, 3 * 512 * sizeof(float), stream);

    const size_t ldsF = (size_t)(4 * N_ + 1024) * sizeof(float) + 1024 * sizeof(int);
    fps_kernel<<<B_, 1024, ldsF, stream>>>(xyz, points, ws_newxyz, ws_newfeat,
                                           out, out_fps);

    ballquery_kernel<<<(B_ * S_) / 8, 256, 0, stream>>>(xyz, ws_newxyz, ws_idx);

    const float invR = 1.0f / (float)R_;

    const size_t lds0 = (size_t)(128 * 133 + 132 * 64 + 128) * sizeof(float)
                      + 128 * sizeof(int);
    gemm0_kernel<<<R_ / 128, 256, lds0, stream>>>(
        xyz, points, ws_newxyz, ws_newfeat, ws_idx, W0, ws_f1, ws_st0);
    stats_finalize_kernel<<<1, 128, 0, stream>>>(ws_st0, 64, invR);

    const size_t lds1 = (size_t)(128 * 65 + 64 * 64 + 2 * 64) * sizeof(float);
    gemm_mid_kernel<64, 64><<<R_ / 128, 256, lds1, stream>>>(
        ws_f1, W1, g0, b0, ws_st0 + 256, ws_st0 + 384, ws_f2, ws_st1);
    stats_finalize_kernel<<<1, 128, 0, stream>>>(ws_st1, 64, invR);

    const size_t lds2 = (size_t)(128 * 65 + 64 * 128 + 2 * 128) * sizeof(float);
    gemm_mid_kernel<64, 128><<<R_ / 128, 256, lds2, stream>>>(
        ws_f2, W2, g1, b1, ws_st1 + 256, ws_st1 + 384, ws_f3, ws_st2);
    stats_finalize_kernel<<<1, 128, 0, stream>>>(ws_st2, 128, invR);

    maxpool_kernel<<<B_ * S_, 128, 0, stream>>>(
        ws_f3, ws_st2 + 256, ws_st2 + 384, g2, b2, out_np);
}